// Cluster_1417339207717
// MI455X (gfx1250) — compile-verified
//
#include <hip/hip_runtime.h>
#include <math.h>

// ---------------------------------------------------------------------------
// CDNA5 (gfx1250) fp32 WMMA implementation of the clustered-attention block.
// GEMM stages use V_WMMA_F32_16X16X4_F32 (wave32, 16x16 tile, K-step 4).
// Block-uniform 64x64 LDS tiles are staged with the Tensor Data Mover
// (tensor_load_to_lds + s_wait_tensorcnt), using TDM row padding to produce
// the pitch-65 layout that avoids LDS bank conflicts on fragment reads.
// ---------------------------------------------------------------------------

typedef float v2f __attribute__((ext_vector_type(2)));
typedef float v8f __attribute__((ext_vector_type(8)));
typedef unsigned int u32;
typedef unsigned int v4u __attribute__((ext_vector_type(4)));
typedef unsigned int v8u __attribute__((ext_vector_type(8)));

static __device__ __forceinline__ v8f wmma_f32_k4(v2f a, v2f b, v8f c) {
  // 8 args: (neg_a, A, neg_b, B, c_mod, C, reuse_a, reuse_b)
  return __builtin_amdgcn_wmma_f32_16x16x4_f32(false, a, false, b, (short)0, c,
                                               false, false);
}

// Raw LDS byte offset of a __shared__ object (generic -> addrspace(3)).
static __device__ __forceinline__ u32 lds_off_of(const void* p) {
  return (u32)(unsigned long long)(const __attribute__((address_space(3))) char*)p;
}

// ---------------------------------------------------------------------------
// TDM: load a 64-row x 64-dword 2D tile from global into LDS with +1 dword of
// padding per row (LDS pitch = 65 dwords). D# groups 0/1 per CDNA5 ISA ch.8:
//   g0 = {count=1, lds_addr, global_addr[56:0], type=2}
//   g1 = {wg_mask=0 (not in cluster), data_size=4B, pad_enable=1,
//         pad_interval=5 (64 dw), pad_amount=0 (1 dw),
//         tensor_dim0, tensor_dim1=64, tile_dim0=64, tile_dim1=64,
//         tensor_dim0_stride}
// Issued by one wave; completion tracked with TENSORcnt.
// ---------------------------------------------------------------------------
static __device__ __forceinline__ void tdm_load_tile64(u32 lds_byte_off,
                                                       const float* gsrc,
                                                       u32 tensor_w_elts,
                                                       u32 row_stride_elts) {
  const unsigned long long ga = (unsigned long long)gsrc;
  v4u g0;
  g0[0] = 0x1u;                                            // count=1, user D#
  g0[1] = lds_byte_off;                                    // lds_addr
  g0[2] = (u32)ga;                                         // global_addr lo32
  g0[3] = (u32)((ga >> 32) & 0x01ffffffu) | 0x80000000u;   // addr[56:32]|type=2
  v8u g1;
  g1[0] = (2u << 16) | (1u << 20) | (5u << 22);  // 4B elts, pad 1dw per 64dw
  g1[1] = tensor_w_elts << 16;                   // tensor_dim0 @ bit 48
  g1[2] = (tensor_w_elts >> 16) | (64u << 16);   // tensor_dim0 hi | tensor_dim1
  g1[3] = (64u << 16);                           // tile_dim0 = 64
  g1[4] = 64u;                                   // tile_dim1 = 64 (tile_dim2=0)
  g1[5] = row_stride_elts;                       // tensor_dim0_stride lo32
  g1[6] = 0u;
  g1[7] = 0u;
  asm volatile("tensor_load_to_lds %0, %1" ::"s"(g0), "s"(g1) : "memory");
}

constexpr int CB   = 16;    // batch
constexpr int CDIM = 256;   // input channels
constexpr int COUT = 256;   // output channels
constexpr int CE   = 8;     // heads (experts)
constexpr int CHD  = 64;    // head dim
constexpr int CP   = 64;    // centers
constexpr int CN   = 4096;  // spatial tokens 64*64
constexpr int CHDE = 512;   // E*HD
constexpr int CBE  = 128;   // B*E
constexpr float CEPS = 1e-12f;

// ---------------------------------------------------------------------------
// Kernel A: bilinear sample cf[b,p,:] + center/value projections + row-norm.
// cn_out  [be][p][h] = normalize_h(cf @ Wfc + bfc)   (per e-group of 64)
// vcent   [be][p][h] =             cf @ Wvc + bvc
// ---------------------------------------------------------------------------
__global__ void k_center(const float* __restrict__ x, const float* __restrict__ ref,
                         const float* __restrict__ Wfc, const float* __restrict__ bfc,
                         const float* __restrict__ Wvc, const float* __restrict__ bvc,
                         float* __restrict__ cn_out, float* __restrict__ vcent_out) {
  __shared__ float cf_s[CDIM];
  __shared__ float cen_s[CHDE];
  __shared__ float vce_s[CHDE];
  __shared__ float nrm_s[CE];
  const int bp = blockIdx.x;
  const int b = bp / CP, p = bp % CP;
  const int t = threadIdx.x;

  const float rx = ref[(b * CP + p) * 2 + 0];
  const float ry = ref[(b * CP + p) * 2 + 1];
  const float ix = (rx + 1.f) * 0.5f * 63.f;
  const float iy = (ry + 1.f) * 0.5f * 63.f;
  const float x0 = floorf(ix), y0 = floorf(iy);
  const float wx = ix - x0, wy = iy - y0;
  const int x0i = min(max((int)x0, 0), 63);
  const int x1i = min(x0i + 1, 63);
  const int y0i = min(max((int)y0, 0), 63);
  const int y1i = min(y0i + 1, 63);

  // channel c == thread id (DIM == blockDim)
  const float* xb = x + ((size_t)b * CDIM + t) * CN;
  const float g00 = xb[y0i * 64 + x0i];
  const float g01 = xb[y0i * 64 + x1i];
  const float g10 = xb[y1i * 64 + x0i];
  const float g11 = xb[y1i * 64 + x1i];
  cf_s[t] = g00 * (1.f - wy) * (1.f - wx) + g01 * (1.f - wy) * wx +
            g10 * wy * (1.f - wx) + g11 * wy * wx;
  __syncthreads();

  for (int o = t; o < CHDE; o += 256) {
    float sc = bfc[o], sv = bvc[o];
    for (int c = 0; c < CDIM; ++c) {
      const float cfv = cf_s[c];
      sc += cfv * Wfc[(size_t)c * CHDE + o];
      sv += cfv * Wvc[(size_t)c * CHDE + o];
    }
    cen_s[o] = sc;
    vce_s[o] = sv;
  }
  __syncthreads();

  if (t < CE) {
    float s = 0.f;
    for (int h = 0; h < CHD; ++h) {
      const float vv = cen_s[t * CHD + h];
      s += vv * vv;
    }
    nrm_s[t] = sqrtf(s) + CEPS;
  }
  __syncthreads();

  for (int o = t; o < CHDE; o += 256) {
    const int e = o >> 6, h = o & 63;
    const size_t idx = ((size_t)(b * CE + e) * CP + p) * CHD + h;
    cn_out[idx] = cen_s[o] / nrm_s[e];
    vcent_out[idx] = vce_s[o];
  }
}

// ---------------------------------------------------------------------------
// Generic 1x1-conv GEMM: Y[b,m,n] = sum_k Wm[m,k] * X[b,k,n] + bias[m]
// 8 waves/block; wave -> one 16x16 tile; block covers 16(M) x 128(N).
// A (weights) staged in LDS with +1 pad; B streamed coalesced from global.
// ---------------------------------------------------------------------------
template <int K>
__global__ void k_gemm1x1(const float* __restrict__ Wm, const float* __restrict__ xin,
                          const float* __restrict__ bias, float* __restrict__ yout,
                          int M) {
  __shared__ float As[16 * (K + 1)];
  const int b = blockIdx.z;
  const int m0 = blockIdx.y * 16;
  const int wid = threadIdx.x >> 5, lane = threadIdx.x & 31;
  const int n0 = blockIdx.x * 128 + wid * 16;
  const float* xb = xin + (size_t)b * K * CN;
  float* yb = yout + (size_t)b * M * CN;

  for (int idx = threadIdx.x; idx < 16 * K; idx += 256) {
    const int r = idx / K, c = idx % K;
    As[r * (K + 1) + c] = Wm[(size_t)(m0 + r) * K + c];
  }
  __syncthreads();

  const int mrow = lane & 15;
  const int g2 = (lane >> 4) * 2;
  v8f acc = {0.f, 0.f, 0.f, 0.f, 0.f, 0.f, 0.f, 0.f};
#pragma unroll 4
  for (int k0 = 0; k0 < K; k0 += 4) {
    v2f a, bb;
    a.x = As[mrow * (K + 1) + k0 + g2];
    a.y = As[mrow * (K + 1) + k0 + g2 + 1];
    const float* bp0 = xb + (size_t)(k0 + g2) * CN + n0 + mrow;
    bb.x = bp0[0];
    bb.y = bp0[CN];
    if (k0 + 16 < K)
      __builtin_prefetch(xb + (size_t)(k0 + 16) * CN + n0 + mrow, 0, 0);
    acc = wmma_f32_k4(a, bb, acc);
  }
  const int mr = (lane >> 4) * 8;
#pragma unroll
  for (int i = 0; i < 8; ++i) {
    const int m = m0 + i + mr;
    yb[(size_t)m * CN + n0 + mrow] = acc[i] + bias[m];
  }
}

// ---------------------------------------------------------------------------
// Kernel C: sim[p,n] = sigmoid(beta + alpha * (cn[p]·f[n]) / (||f[n]||+eps)),
// then hard top-1 over p (masking losers to 0) + per-center counts.
// cn tile staged by TDM (overlapped with the column-norm pass);
// G = cn(64x64) @ ftile(64x64) via WMMA.
// ---------------------------------------------------------------------------
__global__ void k_sim(const float* __restrict__ f, const float* __restrict__ cn,
                      const float* __restrict__ alpha_p, const float* __restrict__ beta_p,
                      float* __restrict__ sim, float* __restrict__ counts) {
  __shared__ float cn_s[64 * 65];
  __shared__ float G_s[64 * 65];
  __shared__ float redv[256];
  __shared__ int redi[256];
  __shared__ float scale_s[64];
  __shared__ int pwin_s[64];

  const int be = blockIdx.y;
  const int b = be / CE, e = be % CE;
  const int n0 = blockIdx.x * 64;
  const int t = threadIdx.x;
  const int wid = t >> 5, lane = t & 31;
  const float* fb = f + (size_t)b * CHDE * CN + (size_t)e * CHD * CN;  // [h][n]
  const float* cb = cn + (size_t)be * CP * CHD;                        // [p][h]

  if (wid == 0)  // wave 0 kicks the DMA; overlaps with the norm pass below
    tdm_load_tile64(lds_off_of(cn_s), cb, CHD, CHD);

  const int nloc = t & 63, part = t >> 6;
  {
    float s = 0.f;
    for (int h = part * 16; h < part * 16 + 16; ++h) {
      const float vv = fb[(size_t)h * CN + n0 + nloc];
      s += vv * vv;
    }
    redv[t] = s;
  }
  if (wid == 0) __builtin_amdgcn_s_wait_tensorcnt(0);
  __syncthreads();
  if (part == 0) {
    const float s = redv[nloc] + redv[64 + nloc] + redv[128 + nloc] + redv[192 + nloc];
    scale_s[nloc] = 1.f / (sqrtf(s) + CEPS);
  }

  const int mrow = lane & 15, g2 = (lane >> 4) * 2, mr = (lane >> 4) * 8;
  for (int it = 0; it < 2; ++it) {
    const int tt = wid + it * 8;
    const int mt = tt & 3, nt = tt >> 2;
    v8f acc = {0.f, 0.f, 0.f, 0.f, 0.f, 0.f, 0.f, 0.f};
#pragma unroll
    for (int k0 = 0; k0 < CHD; k0 += 4) {
      v2f a, bb;
      a.x = cn_s[(mt * 16 + mrow) * 65 + k0 + g2];
      a.y = cn_s[(mt * 16 + mrow) * 65 + k0 + g2 + 1];
      const float* bp0 = fb + (size_t)(k0 + g2) * CN + n0 + nt * 16 + mrow;
      bb.x = bp0[0];
      bb.y = bp0[CN];
      acc = wmma_f32_k4(a, bb, acc);
    }
#pragma unroll
    for (int i = 0; i < 8; ++i)
      G_s[(mt * 16 + i + mr) * 65 + nt * 16 + mrow] = acc[i];
  }
  __syncthreads();

  const float alpha = alpha_p[0], beta = beta_p[0];
  {
    float mx = -3.4028235e38f;
    int ai = 0;
    const float sc = scale_s[nloc];
    for (int p = part * 16; p < part * 16 + 16; ++p) {
      const float s = beta + alpha * G_s[p * 65 + nloc] * sc;
      if (s > mx) { mx = s; ai = p; }
    }
    redv[t] = mx;
    redi[t] = ai;
  }
  __syncthreads();
  if (part == 0) {
    float mx = redv[nloc];
    int ai = redi[nloc];
    for (int q = 1; q < 4; ++q) {
      const float vq = redv[q * 64 + nloc];
      if (vq > mx) { mx = vq; ai = redi[q * 64 + nloc]; }
    }
    pwin_s[nloc] = ai;
    atomicAdd(&counts[be * CP + ai], 1.f);  // exact integer fp adds: deterministic
  }
  __syncthreads();
  {
    const float sc = scale_s[nloc];
    const int pw = pwin_s[nloc];
    float* so = sim + (size_t)be * CP * CN + n0 + nloc;
    for (int p = part * 16; p < part * 16 + 16; ++p) {
      const float s = beta + alpha * G_s[p * 65 + nloc] * sc;
      const float val = (p == pw) ? (1.f / (1.f + __expf(-s))) : 0.f;
      so[(size_t)p * CN] = val;
    }
  }
}

// ---------------------------------------------------------------------------
// Kernel D: outp[be,p,c] = (sum_n sim[p,n]*v[b,e*64+c,n] + vcent) / (count+1)
// One block per be; K=4096 chunked by 64. Both operands are lane-strided in
// global, so each 64x64 chunk is staged to padded LDS via TDM (no VGPR
// round-trip, no VALU address math in the staging path).
// ---------------------------------------------------------------------------
__global__ void k_agg(const float* __restrict__ sim, const float* __restrict__ v,
                      const float* __restrict__ vcent, const float* __restrict__ counts,
                      float* __restrict__ outp) {
  __shared__ float sim_s[64 * 65];
  __shared__ float v_s[64 * 65];
  const int be = blockIdx.x;
  const int b = be / CE, e = be % CE;
  const int t = threadIdx.x;
  const int wid = t >> 5, lane = t & 31;
  const int mrow = lane & 15, g2 = (lane >> 4) * 2, mr = (lane >> 4) * 8;
  const float* sb = sim + (size_t)be * CP * CN;
  const float* vb = v + (size_t)b * CHDE * CN + (size_t)e * CHD * CN;
  const u32 sim_lds = lds_off_of(sim_s);
  const u32 v_lds = lds_off_of(v_s);

  const int tt0 = wid, tt1 = wid + 8;
  const int mt0 = tt0 & 3, ct0 = tt0 >> 2;
  const int mt1 = tt1 & 3, ct1 = tt1 >> 2;
  v8f acc0 = {0.f, 0.f, 0.f, 0.f, 0.f, 0.f, 0.f, 0.f};
  v8f acc1 = {0.f, 0.f, 0.f, 0.f, 0.f, 0.f, 0.f, 0.f};

  for (int kc = 0; kc < CN; kc += 64) {
    __syncthreads();  // prior chunk fully consumed
    if (wid == 0) {
      tdm_load_tile64(sim_lds, sb + kc, (u32)(CN - kc), CN);
      tdm_load_tile64(v_lds, vb + kc, (u32)(CN - kc), CN);
      __builtin_amdgcn_s_wait_tensorcnt(0);
    }
    __syncthreads();  // publish LDS tiles to all 8 waves
#pragma unroll
    for (int kk = 0; kk < 64; kk += 4) {
      v2f a, bb;
      a.x = sim_s[(mt0 * 16 + mrow) * 65 + kk + g2];
      a.y = sim_s[(mt0 * 16 + mrow) * 65 + kk + g2 + 1];
      bb.x = v_s[(ct0 * 16 + mrow) * 65 + kk + g2];
      bb.y = v_s[(ct0 * 16 + mrow) * 65 + kk + g2 + 1];
      acc0 = wmma_f32_k4(a, bb, acc0);
      a.x = sim_s[(mt1 * 16 + mrow) * 65 + kk + g2];
      a.y = sim_s[(mt1 * 16 + mrow) * 65 + kk + g2 + 1];
      bb.x = v_s[(ct1 * 16 + mrow) * 65 + kk + g2];
      bb.y = v_s[(ct1 * 16 + mrow) * 65 + kk + g2 + 1];
      acc1 = wmma_f32_k4(a, bb, acc1);
    }
  }
#pragma unroll
  for (int i = 0; i < 8; ++i) {
    {
      const int p = mt0 * 16 + i + mr, c = ct0 * 16 + mrow;
      const size_t idx = ((size_t)be * CP + p) * CHD + c;
      outp[idx] = (acc0[i] + vcent[idx]) / (counts[be * CP + p] + 1.f);
    }
    {
      const int p = mt1 * 16 + i + mr, c = ct1 * 16 + mrow;
      const size_t idx = ((size_t)be * CP + p) * CHD + c;
      outp[idx] = (acc1[i] + vcent[idx]) / (counts[be * CP + p] + 1.f);
    }
  }
}

// ---------------------------------------------------------------------------
// Kernel E: out_img[b,e*64+c,n] = sum_p sim[p,n]*outp[p,c]  (dispatch).
// Computed transposed (M=c, N=n, K=p) so B-loads and D-stores are coalesced
// along n; outp tile staged via TDM.
// ---------------------------------------------------------------------------
__global__ void k_disp(const float* __restrict__ sim, const float* __restrict__ outp,
                       float* __restrict__ outimg) {
  __shared__ float op_s[64 * 65];  // [p][c]
  const int be = blockIdx.y;
  const int b = be / CE, e = be % CE;
  const int t = threadIdx.x;
  const int wid = t >> 5, lane = t & 31;
  if (wid == 0) {
    tdm_load_tile64(lds_off_of(op_s), outp + (size_t)be * CP * CHD, CHD, CHD);
    __builtin_amdgcn_s_wait_tensorcnt(0);
  }
  __syncthreads();

  const int mrow = lane & 15, g2 = (lane >> 4) * 2, mr = (lane >> 4) * 8;
  const int n0 = blockIdx.x * 128 + wid * 16;
  const float* sb = sim + (size_t)be * CP * CN + n0 + mrow;
  float* ob = outimg + (size_t)b * CHDE * CN + (size_t)e * CHD * CN;

  for (int mt = 0; mt < 4; ++mt) {
    v8f acc = {0.f, 0.f, 0.f, 0.f, 0.f, 0.f, 0.f, 0.f};
#pragma unroll
    for (int k0 = 0; k0 < CP; k0 += 4) {
      v2f a, bb;
      a.x = op_s[(k0 + g2) * 65 + mt * 16 + mrow];  // A[c,k] = outp[k][c]
      a.y = op_s[(k0 + g2 + 1) * 65 + mt * 16 + mrow];
      bb.x = sb[(size_t)(k0 + g2) * CN];
      bb.y = sb[(size_t)(k0 + g2 + 1) * CN];
      acc = wmma_f32_k4(a, bb, acc);
    }
#pragma unroll
    for (int i = 0; i < 8; ++i) {
      const int c = mt * 16 + i + mr;
      ob[(size_t)c * CN + n0 + mrow] = acc[i];
    }
  }
}

// ---------------------------------------------------------------------------
// Kernel G: c_fin[b,p,:] = c_out[b,p,:] @ Wpc + bpc   (c_out gathered from outp)
// ---------------------------------------------------------------------------
__global__ void k_cfin(const float* __restrict__ outp, const float* __restrict__ Wpc,
                       const float* __restrict__ bpc, float* __restrict__ cfin) {
  __shared__ float row_s[CHDE];
  const int bp = blockIdx.x;
  const int b = bp / CP, p = bp % CP;
  const int t = threadIdx.x;
  for (int i = t; i < CHDE; i += 256) {
    const int e = i >> 6, h = i & 63;
    row_s[i] = outp[((size_t)(b * CE + e) * CP + p) * CHD + h];
  }
  __syncthreads();
  float s = bpc[t];
  for (int c = 0; c < CHDE; ++c) s += row_s[c] * Wpc[(size_t)c * COUT + t];
  cfin[((size_t)(b * CP) + p) * COUT + t] = s;
}

// ---------------------------------------------------------------------------
extern "C" void kernel_launch(void* const* d_in, const int* in_sizes, int n_in,
                              void* d_out, int out_size, void* d_ws, size_t ws_size,
                              hipStream_t stream) {
  (void)in_sizes; (void)n_in; (void)out_size; (void)ws_size;
  const float* x     = (const float*)d_in[0];
  const float* ref   = (const float*)d_in[1];
  const float* Wf    = (const float*)d_in[2];
  const float* bf    = (const float*)d_in[3];
  const float* Wfc   = (const float*)d_in[4];
  const float* bfc   = (const float*)d_in[5];
  const float* Wv    = (const float*)d_in[6];
  const float* bv    = (const float*)d_in[7];
  const float* Wvc   = (const float*)d_in[8];
  const float* bvc   = (const float*)d_in[9];
  const float* Wp    = (const float*)d_in[10];
  const float* bp    = (const float*)d_in[11];
  const float* Wpc   = (const float*)d_in[12];
  const float* bpc   = (const float*)d_in[13];
  const float* alpha = (const float*)d_in[14];
  const float* beta  = (const float*)d_in[15];

  // Workspace layout (floats). out_img aliases f (f dead after k_sim).
  float* ws     = (float*)d_ws;
  float* cn     = ws;                                   // 128*64*64
  float* vcent  = cn + (size_t)CBE * CP * CHD;          // 128*64*64
  float* outp   = vcent + (size_t)CBE * CP * CHD;       // 128*64*64
  float* counts = outp + (size_t)CBE * CP * CHD;        // 128*64
  float* f      = counts + (size_t)CBE * CP;            // 16*512*4096
  float* v      = f + (size_t)CB * CHDE * CN;           // 16*512*4096
  float* sim    = v + (size_t)CB * CHDE * CN;           // 128*64*4096
  float* outimg = f;                                    // alias (f consumed by k_sim)

  float* out_main = (float*)d_out;                      // [16,256,64,64]
  float* cfin = out_main + (size_t)CB * COUT * CN;      // [16,64,256]

  hipMemsetAsync(counts, 0, (size_t)CBE * CP * sizeof(float), stream);

  k_center<<<dim3(CB * CP), 256, 0, stream>>>(x, ref, Wfc, bfc, Wvc, bvc, cn, vcent);
  k_gemm1x1<CDIM><<<dim3(CN / 128, CHDE / 16, CB), 256, 0, stream>>>(Wf, x, bf, f, CHDE);
  k_gemm1x1<CDIM><<<dim3(CN / 128, CHDE / 16, CB), 256, 0, stream>>>(Wv, x, bv, v, CHDE);
  k_sim<<<dim3(CN / 64, CBE), 256, 0, stream>>>(f, cn, alpha, beta, sim, counts);
  k_agg<<<dim3(CBE), 256, 0, stream>>>(sim, v, vcent, counts, outp);
  k_disp<<<dim3(CN / 128, CBE), 256, 0, stream>>>(sim, outp, outimg);
  k_gemm1x1<CHDE><<<dim3(CN / 128, COUT / 16, CB), 256, 0, stream>>>(Wp, outimg, bp, out_main, COUT);
  k_cfin<<<dim3(CB * CP), 256, 0, stream>>>(outp, Wpc, bpc, cfin);
}